// Scatter_40656160424517
// MI455X (gfx1250) — compile-verified
//
#include <hip/hip_runtime.h>
#include <hip/hip_bf16.h>

typedef float v2f __attribute__((ext_vector_type(2)));
typedef float v4f __attribute__((ext_vector_type(4)));
typedef float v8f __attribute__((ext_vector_type(8)));
typedef int   v4i __attribute__((ext_vector_type(4)));

#define CH 64
#define LDS_PITCH 66  // 64 + 2 floats pad -> lanes 0..15 hit distinct banks

// ---------------------------------------------------------------------------
// Zero-init the output (harness poisons d_out with 0xAA before timing).
// ---------------------------------------------------------------------------
__global__ __launch_bounds__(256) void zero_out_kernel(float* __restrict__ out,
                                                       int n_float4) {
    int i = blockIdx.x * blockDim.x + threadIdx.x;
    if (i < n_float4) {
        v4f z = {};
        *(v4f*)(out + (size_t)i * 4) = z;
    }
}

// ---------------------------------------------------------------------------
// Fused gather-GEMM-scatter:
//   per 16-row tile (one wave): A tile staged via CDNA5 async global->LDS
//   (ASYNCcnt path, no VGPR round-trip), y = A_tile @ W^T + bias via WMMA
//   f32 16x16x4, then atomic scatter-add into out[idx[row]].
// ---------------------------------------------------------------------------
__global__ __launch_bounds__(256) void fused_gemm_scatter_kernel(
    const float* __restrict__ A,     // [nRows, 64]
    const float* __restrict__ W,     // [64, 64] row = out_ch, col = in_ch
    const float* __restrict__ bias,  // [64]
    const int* __restrict__ idx,     // [nRows]
    float* out,                      // [NUM_SEG, 64]
    int nRows) {
    __shared__ float ldsA[8 * 16 * LDS_PITCH];  // per-wave 16x64 A tile, padded
    __shared__ int ldsIdx[8 * 16];               // per-wave 16 segment ids

    const int wave = threadIdx.x >> 5;
    const int lane = threadIdx.x & 31;
    const int hi = lane >> 4;   // 0: lanes 0-15, 1: lanes 16-31
    const int lm = lane & 15;

    float* aT = &ldsA[wave * 16 * LDS_PITCH];
    int* iT = &ldsIdx[wave * 16];

    const long tile = (long)blockIdx.x * 8 + wave;  // 16-row tile index
    const long rowBase = tile * 16;
    const bool valid = (rowBase + 16) <= (long)nRows;

    if (valid) {
        // Async-stage the 16x64 A tile straight into LDS (no VGPR round-trip).
        // 8 iters, each lane moves one b128 (4 floats).
        #pragma unroll
        for (int it = 0; it < 8; ++it) {
            const int t = it * 32 + lane;       // 0..255 float4 slots
            const int r = t >> 4;               // row in tile
            const int c4 = (t & 15) << 2;       // starting column
            const float* src = A + (rowBase + r) * CH + c4;
            float* dst = aT + r * LDS_PITCH + c4;
            __builtin_amdgcn_global_load_async_to_lds_b128(
                (__attribute__((address_space(1))) v4i*)src,
                (__attribute__((address_space(3))) v4i*)dst,
                /*imm offset=*/0, /*cpol=*/0);
        }
        if (lane < 16) iT[lane] = idx[rowBase + lane];
    }
    __builtin_amdgcn_s_wait_asynccnt(0);
    __syncthreads();
    if (!valid) return;

    // Loop over the four 16-wide output-column tiles.
    #pragma unroll
    for (int nt = 0; nt < 4; ++nt) {
        // B fragments: B = W^T tile (4x16 per k-step).
        // lane: b.x = W[nt*16+lm][4k + 2*hi], b.y = next input channel.
        v2f bfrag[16];
        const float* wbase = W + (nt * 16 + lm) * CH + hi * 2;
        #pragma unroll
        for (int k = 0; k < 16; ++k)
            bfrag[k] = *(const v2f*)(wbase + k * 4);

        v8f acc = {};
        #pragma unroll
        for (int k = 0; k < 16; ++k) {
            // A fragment from LDS: a.x = A[lm][4k + 2*hi], a.y = next col.
            v2f afrag = *(const v2f*)(aT + lm * LDS_PITCH + k * 4 + hi * 2);
            acc = __builtin_amdgcn_wmma_f32_16x16x4_f32(
                /*neg_a=*/false, afrag, /*neg_b=*/false, bfrag[k],
                /*c_mod=*/(short)0, acc, /*reuse_a=*/false, /*reuse_b=*/false);
        }

        const int col = nt * 16 + lm;
        const float bv = bias[col];

        // D layout: element i -> row (i + 8*hi), col (nt*16 + lm).
        #pragma unroll
        for (int i = 0; i < 8; ++i) {
            const int m = i + hi * 8;
            const int seg = iT[m];
            const float val = acc[i] + bv;
            __hip_atomic_fetch_add(out + (long)seg * CH + col, val,
                                   __ATOMIC_RELAXED, __HIP_MEMORY_SCOPE_AGENT);
        }
    }
}

// ---------------------------------------------------------------------------
// Harness entry point.
// ---------------------------------------------------------------------------
extern "C" void kernel_launch(void* const* d_in, const int* in_sizes, int n_in,
                              void* d_out, int out_size, void* d_ws, size_t ws_size,
                              hipStream_t stream) {
    const float* inputs = (const float*)d_in[0];   // [N, 64]
    const float* weights = (const float*)d_in[1];  // [64, 64]
    const float* bias = (const float*)d_in[2];     // [64]
    const int* idx = (const int*)d_in[3];          // [N]
    float* out = (float*)d_out;                    // [NUM_SEG, 64]

    const int nRows = in_sizes[0] / CH;

    // 1) zero the output accumulator
    const int nF4 = out_size / 4;
    zero_out_kernel<<<(nF4 + 255) / 256, 256, 0, stream>>>(out, nF4);

    // 2) fused GEMM + scatter-add
    const int tiles = (nRows + 15) / 16;     // 16-row tiles
    const int blocks = (tiles + 7) / 8;      // 8 waves (tiles) per block
    fused_gemm_scatter_kernel<<<blocks, 256, 0, stream>>>(
        inputs, weights, bias, idx, out, nRows);
}